// GNN_learner_52475910423111
// MI455X (gfx1250) — compile-verified
//
#include <hip/hip_runtime.h>
#include <hip/hip_bf16.h>

// ---------------------------------------------------------------------------
// GIN (3 layers): scatter-add aggregation + 2x (64x64) GEMM MLP + BatchNorm.
// Memory-bound on the edge gather/scatter (~0.8 GB total, L2-resident on the
// 192MB L2).  Matrix math uses V_WMMA_F32_16X16X4_F32 (full fp32 precision,
// GEMM is ~2.5 GFLOP total -> nowhere near the bottleneck).
// W is staged into LDS k-pair-interleaved so each WMMA B operand is a single
// aligned ds_load_b64 into an even VGPR pair (no re-pack movs).
// ---------------------------------------------------------------------------

typedef __attribute__((ext_vector_type(2))) float v2f;
typedef __attribute__((ext_vector_type(8))) float v8f;

#define DIM 64
#define BN_EPS 1e-5f

// ---------------------------------------------------------------- zero fill
__global__ void zero_f32(float* __restrict__ p, int n) {
    int i = blockIdx.x * blockDim.x + threadIdx.x;
    if (i < n) p[i] = 0.0f;
}

// ------------------------------------------------------------- scatter add
// agg[dst[e]][f] += h[src[e]][f].  64 consecutive threads handle one edge:
// coalesced 256B gather of the source row, coalesced f32 atomics to dest.
__global__ __launch_bounds__(256)
void scatter_add(const float* __restrict__ h,
                 const int*   __restrict__ src,
                 const int*   __restrict__ dst,
                 float*       __restrict__ agg,
                 int E) {
    int idx = blockIdx.x * 256 + threadIdx.x;
    int e = idx >> 6;
    if (e >= E) return;
    int f = idx & 63;
    int s = src[e];
    int d = dst[e];
    atomicAdd(&agg[d * DIM + f], h[s * DIM + f]);
}

// ------------------------------------------------------------- fused GEMM
// out[M,64] = act( A[M,64] @ W[64,64] + bias ), A = A0 (+ A1 when FUSE_ADD).
// One wave -> 16 rows x 64 cols via 4 WMMA f32 16x16x4 accumulators.
// When STATS: per-column sum / sum-of-squares accumulated atomically for BN.
// NOTE: `out` may alias `A1` (tile-local read-then-write; tiles are disjoint)
// so neither carries __restrict__.
template <bool FUSE_ADD, bool RELU_OUT, bool STATS>
__global__ __launch_bounds__(256)
void gin_mlp_gemm(const float* __restrict__ A0,
                  const float* A1,
                  const float* __restrict__ W,
                  const float* __restrict__ bias,
                  float* out,
                  float* __restrict__ stats,   // [0..63]=sum, [64..127]=sumsq
                  int N) {
    // W interleaved by k-pair: sWI[p*WSI + 2n + j] = W[2p+j][n].
    // WSI=160 -> lane halves (pair p vs p+1) offset 160 = 32 mod 64 banks:
    // lower half hits banks {2n,2n+1}, upper {2n+32,..} -> conflict-free b64.
    constexpr int WSI = 160;
    constexpr int AS  = 68;   // A stride: bank (4m+k)%64 unique per lane;
                              // rows stay 16B aligned for float4 staging.
    __shared__ float sWI[32 * WSI];
    __shared__ float sBias[DIM];
    __shared__ float sA[8 * 16 * AS];

    const int tid  = threadIdx.x;
    const int wave = tid >> 5;
    const int lane = tid & 31;
    const int half = lane >> 4;   // 0: lanes 0-15, 1: lanes 16-31
    const int l16  = lane & 15;

    // Stage W (64x64) as 1024 float4s across 256 threads, interleaving pairs.
    for (int i = tid; i < DIM * 16; i += 256) {
        int r = i >> 4, c = (i & 15) << 2;
        const float4 v = *(const float4*)(W + r * DIM + c);
        float* d = &sWI[(r >> 1) * WSI + 2 * c + (r & 1)];
        d[0] = v.x; d[2] = v.y; d[4] = v.z; d[6] = v.w;
    }
    if (tid < DIM) sBias[tid] = bias[tid];

    // Stage this wave's 16x64 A tile (h + agg fused), coalesced float4 loads.
    const int rowBase = blockIdx.x * 128 + wave * 16;
    float* myA = &sA[wave * 16 * AS];
    for (int i = lane; i < 16 * 16; i += 32) {
        int r = i >> 4, c4 = i & 15;
        int row = rowBase + r;
        float4 v = make_float4(0.f, 0.f, 0.f, 0.f);
        if (row < N) {
            v = *(const float4*)(A0 + row * DIM + c4 * 4);
            if (FUSE_ADD) {
                const float4 u = *(const float4*)(A1 + row * DIM + c4 * 4);
                v.x += u.x; v.y += u.y; v.z += u.z; v.w += u.w;
            }
        }
        float* d = &myA[r * AS + c4 * 4];
        d[0] = v.x; d[1] = v.y; d[2] = v.z; d[3] = v.w;
    }
    __syncthreads();

    // K loop: 16 steps of 16x16x4 f32 WMMA for each of 4 N-tiles.
    // A (16x4 f32 ISA layout): lane m holds K = {4t+2*half, +1} -> one v2f.
    // B (4x16 f32): lane -> col l16+16*nt, pair rows from interleaved LDS.
    v8f c0 = {}, c1 = {}, c2 = {}, c3 = {};
    const float* aRow = &myA[l16 * AS + half * 2];
    for (int t = 0; t < 16; ++t) {
        const int k = t * 4;
        const int p = 2 * t + half;           // k-pair index (4t+2*half)/2
        const v2f a = *(const v2f*)(aRow + k);
        const float* wp = &sWI[p * WSI + 2 * l16];
        const v2f b0 = *(const v2f*)(wp);
        const v2f b1 = *(const v2f*)(wp + 32);
        const v2f b2 = *(const v2f*)(wp + 64);
        const v2f b3 = *(const v2f*)(wp + 96);
        c0 = __builtin_amdgcn_wmma_f32_16x16x4_f32(false, a, false, b0,
                                                   (short)0, c0, false, false);
        c1 = __builtin_amdgcn_wmma_f32_16x16x4_f32(false, a, false, b1,
                                                   (short)0, c1, false, false);
        c2 = __builtin_amdgcn_wmma_f32_16x16x4_f32(false, a, false, b2,
                                                   (short)0, c2, false, false);
        c3 = __builtin_amdgcn_wmma_f32_16x16x4_f32(false, a, false, b3,
                                                   (short)0, c3, false, false);
    }

    // Epilogue. C/D layout: lane half -> rows {0..7 | 8..15}, VGPR v = row+v,
    // column = nt*16 + l16.
    const int rBase = rowBase + half * 8;
    v8f accs[4] = {c0, c1, c2, c3};
    for (int nt = 0; nt < 4; ++nt) {
        const int col = nt * 16 + l16;
        const float bv = sBias[col];
        v8f c = accs[nt];
        float s = 0.f, q = 0.f;
#pragma unroll
        for (int v = 0; v < 8; ++v) {
            const int row = rBase + v;
            if (row < N) {
                float d = c[v] + bv;
                if (RELU_OUT) d = fmaxf(d, 0.f);
                out[row * DIM + col] = d;
                if (STATS) { s += d; q += d * d; }
            }
        }
        if (STATS) {
            // Column col lives in lanes L and L^16 -> combine over 16 rows.
            s += __shfl_xor(s, 16);
            q += __shfl_xor(q, 16);
            if (half == 0) {
                atomicAdd(&stats[col], s);
                atomicAdd(&stats[DIM + col], q);
            }
        }
    }
}

// --------------------------------------------------------------- BN apply
__global__ __launch_bounds__(256)
void bn_apply(const float* __restrict__ z,
              const float* __restrict__ stats,
              const float* __restrict__ gamma,
              const float* __restrict__ beta,
              float* __restrict__ out,
              int N, int relu) {
    int i = blockIdx.x * 256 + threadIdx.x;
    if (i >= N * DIM) return;
    int f = i & 63;
    float invN = 1.0f / (float)N;
    float mean = stats[f] * invN;
    float var  = stats[DIM + f] * invN - mean * mean;
    float inv  = rsqrtf(var + BN_EPS);
    float v = (z[i] - mean) * inv * gamma[f] + beta[f];
    if (relu) v = fmaxf(v, 0.f);
    out[i] = v;
}

// ---------------------------------------------------------------- launcher
extern "C" void kernel_launch(void* const* d_in, const int* in_sizes, int n_in,
                              void* d_out, int out_size, void* d_ws, size_t ws_size,
                              hipStream_t stream) {
    const float* x     = (const float*)d_in[0];
    const int*   edges = (const int*)d_in[1];          // [2, E]
    const float* W1    = (const float*)d_in[2];        // [L, 64, 64]
    const float* b1    = (const float*)d_in[3];        // [L, 64]
    const float* W2    = (const float*)d_in[4];
    const float* b2    = (const float*)d_in[5];
    const float* gamma = (const float*)d_in[6];
    const float* beta  = (const float*)d_in[7];
    float* outp = (float*)d_out;

    const int N = in_sizes[0] / DIM;
    const int E = in_sizes[1] / 2;
    const int L = in_sizes[3] / DIM;
    const int* src = edges;
    const int* dst = edges + E;

    // Workspace: [ agg/z1 | z2 | hbuf | stats(128) ]
    float* agg   = (float*)d_ws;
    float* z2    = agg + (size_t)N * DIM;
    float* hbuf  = z2 + (size_t)N * DIM;
    float* stats = hbuf + (size_t)N * DIM;

    const int ND = N * DIM;
    const dim3 blk(256);
    const dim3 gElem((ND + 255) / 256);
    const dim3 gScat(((size_t)E * DIM + 255) / 256);
    const dim3 gGemm((N + 127) / 128);

    for (int i = 0; i < L; ++i) {
        const float* h_in = (i == 0) ? x : hbuf;
        float* bn_out = (i < L - 1) ? hbuf : outp;

        zero_f32<<<gElem, blk, 0, stream>>>(agg, ND);
        zero_f32<<<dim3(1), blk, 0, stream>>>(stats, 2 * DIM);

        scatter_add<<<gScat, blk, 0, stream>>>(h_in, src, dst, agg, E);

        // z1 = relu((h+agg) @ W1 + b1), written in-place over agg
        gin_mlp_gemm<true, true, false><<<gGemm, blk, 0, stream>>>(
            h_in, agg, W1 + (size_t)i * DIM * DIM, b1 + (size_t)i * DIM,
            agg, stats, N);

        // z2 = z1 @ W2 + b2, fused BN statistics
        gin_mlp_gemm<false, false, true><<<gGemm, blk, 0, stream>>>(
            agg, agg, W2 + (size_t)i * DIM * DIM, b2 + (size_t)i * DIM,
            z2, stats, N);

        bn_apply<<<gElem, blk, 0, stream>>>(
            z2, stats, gamma + (size_t)i * DIM, beta + (size_t)i * DIM,
            bn_out, N, (i < L - 1) ? 1 : 0);
    }
    (void)n_in; (void)out_size; (void)ws_size;
}